// TRCategorical_86964497809537
// MI455X (gfx1250) — compile-verified
//
#include <hip/hip_runtime.h>
#include <math.h>

// TRCategorical.log_prob on gfx1250 (MI455X).
// M=16 modes, D=256 support, R=16 rank, N samples (in_sizes[0]/16).
//
// Design: one wave32 == one sample; 16 sequential 16x16 f32 matmuls as
// 4x V_WMMA_F32_16X16X4_F32 each. The reference's per-step max-rescale is
// replaced by a constant 2^8 scale folded into the precomputed cores table:
//   log(trace(prod G)) - log(trace(prod bars))
// is invariant to any uniform per-step scale, and since E[row sum] of a core
// is 16*softplus(theta) ~ 2^-8, scaling the table by 256 keeps prob ~O(1)
// (drift over 16 steps ~ e^+-0.1, far inside f32 range). The same scaled
// table feeds the normalizer chain (with a 1/256 compensation inside the
// tiny chain so it too stays O(1)), so the scale cancels exactly.
// This removes 16x (5 ds_bpermute + s_wait_dscnt + v_div + v_log) from the
// per-sample critical path.
//
// Data movement: cores table (4 MB) is L2-resident (192 MB L2); each wave
// streams 16 KB of gathered tiles (2 GB total L2->WGP), prefetched up front
// via global_prefetch since all 16 indices are known before the chain runs.
// HBM traffic is only ~12.5 MB total.

typedef float v8f __attribute__((ext_vector_type(8)));
typedef float v2f __attribute__((ext_vector_type(2)));

#define M_MODES 16
#define D_SUP   256
#define R_RANK  16
#define TILE_ELEMS (R_RANK * R_RANK)   // 256 floats = 1KB per core tile
#define WAVES_PER_BLOCK 8
#define SCALE_C 256.0f                  // 2^8 folded into cores table
#define LOGC_TOTAL (128.0f * 0.6931471805599453f)  // 16 * log(256)

__device__ __forceinline__ float softplus_f(float x) {
    if (x > 20.0f)  return x;
    if (x < -20.0f) return expf(x);
    return log1pf(expf(x));
}

// ---------------------------------------------------------------------------
// Kernel A: cores = 256 * softplus(log_cores)   [M*D*R*R elements]
// ---------------------------------------------------------------------------
__global__ void tr_softplus_kernel(const float* __restrict__ in,
                                   float* __restrict__ out, int n) {
    int i = blockIdx.x * blockDim.x + threadIdx.x;
    if (i < n) out[i] = SCALE_C * softplus_f(in[i]);
}

// ---------------------------------------------------------------------------
// Kernel B: bars[m][i][j] = sum_d cores[m][d][i][j]  (scaled; 16 x 256 thr)
// ---------------------------------------------------------------------------
__global__ void tr_bars_kernel(const float* __restrict__ cores,
                               float* __restrict__ bars) {
    int m = blockIdx.x;
    int t = threadIdx.x;
    const float* base = cores + (size_t)m * D_SUP * TILE_ELEMS + t;
    float s = 0.0f;
    for (int d = 0; d < D_SUP; ++d) s += base[(size_t)d * TILE_ELEMS];
    bars[m * TILE_ELEMS + t] = s;
}

// ---------------------------------------------------------------------------
// Kernel C: lognorm = log(trace(prod_m bars_m/256)) + 16*log(256)
// (1 block x 256 threads; dividing out the table scale keeps the chain O(1))
// ---------------------------------------------------------------------------
__global__ void tr_lognorm_kernel(const float* __restrict__ bars,
                                  float* __restrict__ lognorm) {
    __shared__ float cur[TILE_ELEMS];
    __shared__ float nxt[TILE_ELEMS];
    int t = threadIdx.x;
    int i = t >> 4, j = t & 15;
    cur[t] = (i == j) ? 1.0f : 0.0f;
    __syncthreads();
    const float invc = 1.0f / SCALE_C;
    for (int m = 0; m < M_MODES; ++m) {
        const float* bm = bars + m * TILE_ELEMS;
        float acc = 0.0f;
        #pragma unroll
        for (int k = 0; k < R_RANK; ++k)
            acc += cur[i * 16 + k] * (bm[k * 16 + j] * invc);
        nxt[t] = acc;
        __syncthreads();
        cur[t] = nxt[t];
        __syncthreads();
    }
    if (t == 0) {
        float tr = 0.0f;
        #pragma unroll
        for (int k = 0; k < R_RANK; ++k) tr += cur[k * 17];
        lognorm[0] = logf(tr) + LOGC_TOTAL;
    }
}

// ---------------------------------------------------------------------------
// Kernel D: main chain. One wave32 per sample.
//   prob = I; for m: prob = prob @ cores_scaled[m, idx[n,m]]
//   out[n] = log(trace(prob)) - lognorm_total
//
// WMMA f32 16x16x4 layouts (wave32):
//   A (16x4):  lane L (ml=L&15, hi=L>>4): M=ml, v0 holds K=2*hi, v1 K=2*hi+1
//   B (4x16):  lane L: N=L&15, v0 holds row 2*hi, v1 row 2*hi+1
//   C/D 16x16: lane L: N=L&15, VGPR r holds M=r+8*hi
// prob round-trips a 1KB per-wave LDS tile to convert D layout -> A layout.
// ---------------------------------------------------------------------------
__global__ __launch_bounds__(256)
void tr_chain_kernel(const int* __restrict__ index,
                     const float* __restrict__ cores,
                     const float* __restrict__ lognorm,
                     float* __restrict__ out, int nsamples) {
    __shared__ float smem[WAVES_PER_BLOCK * TILE_ELEMS];  // 8KB / block

    const int lane = threadIdx.x & 31;
    const int wave = threadIdx.x >> 5;
    const int n    = blockIdx.x * WAVES_PER_BLOCK + wave;
    if (n >= nsamples) return;                 // wave-uniform branch

    // n is wave-uniform: force it into an SGPR so index fetches become SMEM
    // loads (contiguous 64B row) and tile bases become scalar addresses.
    const int nu = __builtin_amdgcn_readfirstlane(n);

    int idxs[M_MODES];
    #pragma unroll
    for (int m = 0; m < M_MODES; ++m)
        idxs[m] = index[(size_t)nu * M_MODES + m];

    float* prob = smem + wave * TILE_ELEMS;
    const int hi  = lane >> 4;
    const int col = lane & 15;
    const int ml  = lane & 15;

    // prefetch all 16 tiles this wave will chain through (1KB each, L2-hot)
    #pragma unroll
    for (int m = 0; m < M_MODES; ++m) {
        const float* tile = cores + (size_t)(m * D_SUP + idxs[m]) * TILE_ELEMS;
        __builtin_prefetch(tile + lane * 8, 0, 0);  // 32 lanes x 32B = 1KB
    }

    // prob = I in [M][N] row-major LDS
    #pragma unroll
    for (int r = 0; r < 8; ++r) {
        int row = r + 8 * hi;
        prob[row * 16 + col] = (row == col) ? 1.0f : 0.0f;
    }

    // per-lane bases: A chunks from LDS, B chunks from global
    const float* pA = prob + ml * 16 + 2 * hi;      // + 4*kk per chunk

    #pragma unroll
    for (int m = 0; m < M_MODES; ++m) {
        const float* tile = cores + (size_t)(m * D_SUP + idxs[m]) * TILE_ELEMS;
        const float* tb = tile + 32 * hi + col;     // row (2*hi), col

        v8f dd = {0.f, 0.f, 0.f, 0.f, 0.f, 0.f, 0.f, 0.f};
        #pragma unroll
        for (int kk = 0; kk < 4; ++kk) {
            v2f a = { pA[4 * kk], pA[4 * kk + 1] };        // ds_load_b64
            v2f b = { tb[kk * 64], tb[kk * 64 + 16] };     // imm-offset loads
            dd = __builtin_amdgcn_wmma_f32_16x16x4_f32(
                     false, a, false, b, (short)0, dd, false, false);
        }
        // store D back to LDS for next mode's A (no rescale needed: the 2^8
        // table scale keeps entries O(1))
        #pragma unroll
        for (int r = 0; r < 8; ++r)
            prob[(r + 8 * hi) * 16 + col] = dd[r];
    }

    // trace of final prob: diagonal lives in lanes 0..15
    float tval = (lane < R_RANK) ? prob[lane * 17] : 0.0f;
    #pragma unroll
    for (int off = 8; off > 0; off >>= 1)
        tval += __shfl_xor(tval, off, 16);

    if (lane == 0)
        out[n] = logf(tval) - lognorm[0];
}

// ---------------------------------------------------------------------------
extern "C" void kernel_launch(void* const* d_in, const int* in_sizes, int n_in,
                              void* d_out, int out_size, void* d_ws, size_t ws_size,
                              hipStream_t stream) {
    const int*   index     = (const int*)d_in[0];    // [N, 16] int32
    const float* log_cores = (const float*)d_in[1];  // [16, 256, 16, 16] f32
    float*       out       = (float*)d_out;          // [N] f32

    const int nsamples = in_sizes[0] / M_MODES;
    const int ncore    = in_sizes[1];                // M*D*R*R = 1048576

    // workspace: cores_scaled (4MB) | bars (16KB) | lognorm (4B)
    float* cores   = (float*)d_ws;
    float* bars    = cores + ncore;
    float* lognorm = bars + M_MODES * TILE_ELEMS;

    tr_softplus_kernel<<<(ncore + 255) / 256, 256, 0, stream>>>(log_cores, cores, ncore);
    tr_bars_kernel<<<M_MODES, TILE_ELEMS, 0, stream>>>(cores, bars);
    tr_lognorm_kernel<<<1, TILE_ELEMS, 0, stream>>>(bars, lognorm);

    const int nblocks = (nsamples + WAVES_PER_BLOCK - 1) / WAVES_PER_BLOCK;
    tr_chain_kernel<<<nblocks, 256, 0, stream>>>(index, cores, lognorm, out, nsamples);
}